// MultiheadAttention_49263274885800
// MI455X (gfx1250) — compile-verified
//
#include <hip/hip_runtime.h>
#include <hip/hip_bf16.h>
#include <math.h>

// ---------------------------------------------------------------------------
// Problem constants (from the reference)
// ---------------------------------------------------------------------------
#define BB 4
#define TT 2048
#define EE 768
#define HH 12
#define DD 64           // head dim
#define MM (BB * TT)    // 8192 rows for the projection GEMMs
#define SCALE 0.125f    // 1/sqrt(64)

typedef float v2f __attribute__((ext_vector_type(2)));
typedef float v4f __attribute__((ext_vector_type(4)));
typedef float v8f __attribute__((ext_vector_type(8)));
typedef int   v4i __attribute__((ext_vector_type(4)));

// CDNA5 fp32 WMMA: D(16x16 f32) = A(16x4 f32) * B(4x16 f32) + C
__device__ __forceinline__ v8f wmma_f32(v2f a, v2f b, v8f c) {
    return __builtin_amdgcn_wmma_f32_16x16x4_f32(
        /*neg_a=*/false, a, /*neg_b=*/false, b,
        /*c_mod=*/(short)0, c, /*reuse_a=*/false, /*reuse_b=*/false);
}

// ---- CDNA5 async global->LDS copy (ASYNCcnt-tracked), with safe fallbacks --
__device__ __forceinline__ void async_copy_b128(const float* gp, float* lp) {
#if __has_builtin(__builtin_amdgcn_global_load_async_to_lds_b128)
    __builtin_amdgcn_global_load_async_to_lds_b128(
        (__attribute__((address_space(1))) v4i*)gp,
        (__attribute__((address_space(3))) v4i*)lp,
        /*offset=*/0, /*cpol=*/0);
#else
    *(v4f*)lp = *(const v4f*)gp;   // synchronous fallback
#endif
}

__device__ __forceinline__ void wait_async0() {
#if __has_builtin(__builtin_amdgcn_global_load_async_to_lds_b128)
#if __has_builtin(__builtin_amdgcn_s_wait_asynccnt)
    __builtin_amdgcn_s_wait_asynccnt(0);
#else
    asm volatile("s_wait_asynccnt 0" ::: "memory");
#endif
#endif
}

// ---------------------------------------------------------------------------
// GEMM + bias: C[8192,768] = A[8192,768] @ W[768,768] + bias  (row-major f32)
// Block (8 waves) owns a 128x64 tile; the full W panel [768 x 64] is staged
// once into LDS via async-to-LDS b128 DMA, then all B-fragments are ds loads.
// ---------------------------------------------------------------------------
#define WPITCH 68    // LDS row pitch (floats): 64 + 4 pad, keeps 16B alignment

__global__ __launch_bounds__(256) void gemm_bias_kernel(
    const float* __restrict__ A, const float* __restrict__ W,
    const float* __restrict__ bias, float* __restrict__ C) {

    __shared__ float Wl[EE * WPITCH];        // 768*68*4 = 204 KB (CDNA5 WGP LDS)

    const int tid  = threadIdx.x;
    const int lane = tid & 31;
    const int w    = tid >> 5;               // wave 0..7
    const int col0 = blockIdx.x * 64;        // N tile
    const int row0 = (blockIdx.y * 8 + w) * 16;

    // ---- stage W[:, col0..col0+63] into LDS (async DMA, no VGPR traffic) ---
    {
        const int c4 = (tid & 15) << 2;      // dword column 0,4,..,60
        for (int k = tid >> 4; k < EE; k += 16) {
            async_copy_b128(W + (size_t)k * EE + col0 + c4, &Wl[k * WPITCH + c4]);
        }
        wait_async0();
        __syncthreads();
    }

    const int m     = lane & 15;             // A row in tile / B column
    const int ksel  = (lane >> 4) << 1;      // lanes 0-15 -> K 0,1 ; 16-31 -> 2,3
    const int rbase = (lane >> 4) << 3;

    v8f acc[4] = {};
    const float* __restrict__ arow = A + (size_t)(row0 + m) * EE;

    for (int k0 = 0; k0 < EE; k0 += 4) {
        v2f a;
        a.x = arow[k0 + ksel];
        a.y = arow[k0 + ksel + 1];
        const float* wr = &Wl[(k0 + ksel) * WPITCH + m];
#pragma unroll
        for (int j = 0; j < 4; ++j) {
            v2f b;
            b.x = wr[j * 16];                // W[k0+ksel  ][col0 + j*16 + m]
            b.y = wr[j * 16 + WPITCH];       // W[k0+ksel+1][col0 + j*16 + m]
            acc[j] = wmma_f32(a, b, acc[j]);
        }
    }

    // C/D layout: VGPR r -> row = r + 8*(lane>>4), col = lane&15
#pragma unroll
    for (int j = 0; j < 4; ++j) {
        const int n  = col0 + j * 16 + m;
        const float bv = bias[n];
#pragma unroll
        for (int r = 0; r < 8; ++r) {
            C[(size_t)(row0 + rbase + r) * EE + n] = acc[j][r] + bv;
        }
    }
}

// ---------------------------------------------------------------------------
// Attention: one workgroup (8 waves) per (b, h, 16-query-row tile).
// Full causal score slab for the 16 rows lives in LDS (320 KB WGP budget).
// ---------------------------------------------------------------------------
#define SPITCH 2052   // 2048 + 4 float pad -> rotate LDS banks per row

__global__ __launch_bounds__(256) void attn_kernel(
    const float* __restrict__ Qp, const float* __restrict__ Kp,
    const float* __restrict__ Vp, float* __restrict__ att,
    float* __restrict__ ypre) {

    __shared__ float S[16][SPITCH];      // 16 x 2048 scores (padded) ~128.25 KB
    __shared__ float part[8][16][DD];    // per-wave partial P*V, 32 KB

    const int qt  = blockIdx.x;          // 0..127 query tile
    const int h   = blockIdx.y;          // 0..11
    const int b   = blockIdx.z;          // 0..3
    const int tid = threadIdx.x;
    const int lane = tid & 31;
    const int w    = tid >> 5;           // wave 0..7
    const int q0   = qt * 16;
    const int m    = lane & 15;
    const int ksel = (lane >> 4) << 1;
    const int rbase = (lane >> 4) << 3;
    const int kmax = (qt + 1) * 16;      // causal frontier (exclusive)

    const float* __restrict__ Qb = Qp + (size_t)b * TT * EE + (size_t)h * DD;
    const float* __restrict__ Kb = Kp + (size_t)b * TT * EE + (size_t)h * DD;
    const float* __restrict__ Vb = Vp + (size_t)b * TT * EE + (size_t)h * DD;

    // ---------------- Phase 1: S = scale * Q Kt, causal mask ---------------
    const float* __restrict__ qrow = Qb + (size_t)(q0 + m) * EE;  // A: Q row
    for (int jt = w; jt <= qt; jt += 8) {
        // B[k][n] = K[jt*16 + n][k]  (n = lane&15)
        const float* __restrict__ krow = Kb + (size_t)(jt * 16 + m) * EE;
        v8f acc = {};
#pragma unroll
        for (int k0 = 0; k0 < DD; k0 += 4) {
            v2f a, bv;
            a.x  = qrow[k0 + ksel];  a.y  = qrow[k0 + ksel + 1];
            bv.x = krow[k0 + ksel];  bv.y = krow[k0 + ksel + 1];
            acc = wmma_f32(a, bv, acc);
        }
        const int col = jt * 16 + m;
#pragma unroll
        for (int r = 0; r < 8; ++r) {
            const int row = rbase + r;                 // 0..15 (query row local)
            float s = acc[r] * SCALE;
            if (col > q0 + row) s = -__builtin_inff(); // causal mask
            S[row][col] = s;
        }
    }
    __syncthreads();

    // ---------------- Phase 2: softmax over each of the 16 rows ------------
    {
        const int row = 2 * w + (lane >> 4);  // wave w owns rows 2w, 2w+1
        const int sub = lane & 15;
        float mx = -__builtin_inff();
        for (int c = sub; c < kmax; c += 16) mx = fmaxf(mx, S[row][c]);
#pragma unroll
        for (int off = 8; off > 0; off >>= 1) mx = fmaxf(mx, __shfl_xor(mx, off, 16));
        float sum = 0.0f;
        for (int c = sub; c < kmax; c += 16) {
            float p = expf(S[row][c] - mx);   // exp(-inf) = 0 above diagonal
            S[row][c] = p;
            sum += p;
        }
#pragma unroll
        for (int off = 8; off > 0; off >>= 1) sum += __shfl_xor(sum, off, 16);
        const float inv = 1.0f / sum;
        const size_t abase = (((size_t)(b * HH + h)) * TT + (q0 + row)) * TT;
        for (int c = sub; c < kmax; c += 16) {
            float p = S[row][c] * inv;
            S[row][c] = p;
            att[abase + c] = p;
        }
    }
    // zeros above the causal frontier in att_weights (b128 store stream)
    {
        const v4f z4 = {0.0f, 0.0f, 0.0f, 0.0f};
        const int nz4 = (TT - kmax) >> 2;              // multiple of 4
        for (int r = 0; r < 16; ++r) {
            v4f* dst = (v4f*)(att + (((size_t)(b * HH + h)) * TT + (q0 + r)) * TT + kmax);
            for (int i = tid; i < nz4; i += 256) dst[i] = z4;
        }
    }
    __syncthreads();

    // ---------------- Phase 3: y = P @ V  (split K over 8 waves) -----------
    v8f yacc[4] = {};
    for (int kt = w; kt <= qt; kt += 8) {
#pragma unroll
        for (int k0 = 0; k0 < 16; k0 += 4) {
            const int kg = kt * 16 + k0 + ksel;
            v2f a;
            a.x = S[m][kg];
            a.y = S[m][kg + 1];
            const float* __restrict__ vrow = Vb + (size_t)kg * EE;
#pragma unroll
            for (int j = 0; j < 4; ++j) {
                const int n = j * 16 + m;
                v2f bv;
                bv.x = vrow[n];
                bv.y = vrow[EE + n];
                yacc[j] = wmma_f32(a, bv, yacc[j]);
            }
        }
    }
#pragma unroll
    for (int j = 0; j < 4; ++j)
#pragma unroll
        for (int r = 0; r < 8; ++r)
            part[w][rbase + r][j * 16 + m] = yacc[j][r];
    __syncthreads();

    // reduce 8 partials -> ypre[b, q0+r, h*64+n]
    for (int idx = tid; idx < 16 * DD; idx += 256) {
        const int r = idx >> 6;
        const int n = idx & 63;
        float s = 0.0f;
#pragma unroll
        for (int ww = 0; ww < 8; ++ww) s += part[ww][r][n];
        ypre[((size_t)b * TT + q0 + r) * EE + (size_t)h * DD + n] = s;
    }
}

// ---------------------------------------------------------------------------
// Launch
// ---------------------------------------------------------------------------
extern "C" void kernel_launch(void* const* d_in, const int* in_sizes, int n_in,
                              void* d_out, int out_size, void* d_ws, size_t ws_size,
                              hipStream_t stream) {
    const float* x  = (const float*)d_in[0];
    const float* Wq = (const float*)d_in[1];
    const float* bq = (const float*)d_in[2];
    const float* Wk = (const float*)d_in[3];
    const float* bk = (const float*)d_in[4];
    const float* Wv = (const float*)d_in[5];
    const float* bv = (const float*)d_in[6];
    const float* Wo = (const float*)d_in[7];
    const float* bo = (const float*)d_in[8];
    // d_in[9] = attn_mask (pure causal -> implemented analytically)

    const size_t BTE = (size_t)BB * TT * EE;
    float* y_out   = (float*)d_out;
    float* att_out = y_out + BTE;

    float* Qd = (float*)d_ws;
    float* Kd = Qd + BTE;
    float* Vd = Kd + BTE;
    float* Yp = Vd + BTE;

    const dim3 ggrid(EE / 64, MM / 128);    // (12, 64) blocks of 256 threads

    gemm_bias_kernel<<<ggrid, 256, 0, stream>>>(x,  Wq, bq, Qd);
    gemm_bias_kernel<<<ggrid, 256, 0, stream>>>(x,  Wk, bk, Kd);
    gemm_bias_kernel<<<ggrid, 256, 0, stream>>>(x,  Wv, bv, Vd);

    attn_kernel<<<dim3(TT / 16, HH, BB), 256, 0, stream>>>(Qd, Kd, Vd, att_out, Yp);

    gemm_bias_kernel<<<ggrid, 256, 0, stream>>>(Yp, Wo, bo, y_out);
}